// Value_47656957116635
// MI455X (gfx1250) — compile-verified
//
#include <hip/hip_runtime.h>
#include <stdint.h>

typedef __attribute__((ext_vector_type(2))) float v2f;
typedef __attribute__((ext_vector_type(8))) float v8f;

#define BN_EPS 1e-5f

static __device__ __forceinline__ void atomAddF(float* p, float v) {
    __hip_atomic_fetch_add(p, v, __ATOMIC_RELAXED, __HIP_MEMORY_SCOPE_AGENT);
}

// ---------------- degrees ----------------
__global__ void deg_kernel(const int* __restrict__ src, const int* __restrict__ dst,
                           float* __restrict__ dout, float* __restrict__ din, int E) {
    int e = blockIdx.x * blockDim.x + threadIdx.x;
    if (e < E) {
        atomAddF(&dout[src[e]], 1.0f);
        atomAddF(&din[dst[e]], 1.0f);
    }
}

__global__ void rsqrt_deg_kernel(float* __restrict__ a, float* __restrict__ b, int N) {
    int i = blockIdx.x * blockDim.x + threadIdx.x;
    if (i < N) {
        a[i] = rsqrtf(fmaxf(a[i], 1.0f));
        b[i] = rsqrtf(fmaxf(b[i], 1.0f));
    }
}

// ---------------- h = x * scale[row], width 256, float4 ----------------
__global__ void rowscale_kernel(const float* __restrict__ x, const float* __restrict__ s,
                                float* __restrict__ y, int nquad /* N*64 */) {
    int i = blockIdx.x * blockDim.x + threadIdx.x;
    if (i < nquad) {
        int node = i >> 6;
        float4 v = ((const float4*)x)[i];
        float sc = s[node];
        v.x *= sc; v.y *= sc; v.z *= sc; v.w *= sc;
        ((float4*)y)[i] = v;
    }
}

// ---------------- agg[dst] += h[src], 64 lanes (float4 each) per edge ----------------
__global__ void aggregate_kernel(const float* __restrict__ h, const int* __restrict__ src,
                                 const int* __restrict__ dst, float* __restrict__ agg, int E) {
    long long t = (long long)blockIdx.x * blockDim.x + threadIdx.x;
    int e = (int)(t >> 6);
    int q = (int)(t & 63);
    if (e < E) {
        int s = src[e];
        int d = dst[e];
        float4 v = ((const float4*)(h + (size_t)s * 256))[q];
        float* p = agg + (size_t)d * 256 + (size_t)q * 4;
        atomAddF(p + 0, v.x);
        atomAddF(p + 1, v.y);
        atomAddF(p + 2, v.z);
        atomAddF(p + 3, v.w);
    }
}

// ---------------- Y = (diag(rowscale) X) @ W + bias ; one wave per 16x16 tile ----------------
// f32 WMMA 16x16x4. A (16x4): lanes 0-15 hold K=0,1; lanes 16-31 hold K=2,3 (contiguous pair).
// B (4x16): lanes 0-15 rows K=0,1; lanes 16-31 rows K=2,3; lane%16 = column.
// C/D: vgpr r -> M = r + (lane<16 ? 0 : 8), N = lane%16.
__global__ void wmma_gemm_kernel(const float* __restrict__ X, const float* __restrict__ W,
                                 const float* __restrict__ bias, const float* __restrict__ rowscale,
                                 float* __restrict__ Y, int Nrows, int K, int Ncols) {
    int wave = blockIdx.x * (blockDim.x >> 5) + (threadIdx.x >> 5);
    int colTiles = Ncols >> 4;
    int rowTiles = Nrows >> 4;
    if (wave >= rowTiles * colTiles) return;
    int rt = wave / colTiles;
    int ct = wave - rt * colTiles;

    int lane = threadIdx.x & 31;
    int half = lane >> 4;     // 0: K pair {0,1}, 1: K pair {2,3}
    int lm   = lane & 15;
    int row  = (rt << 4) + lm;
    int col  = (ct << 4) + lm;

    const float* xr = X + (size_t)row * K;
    v8f acc = {0.f, 0.f, 0.f, 0.f, 0.f, 0.f, 0.f, 0.f};

    for (int k = 0; k < K; k += 4) {
        int kk = k + (half << 1);
        v2f a = *(const v2f*)(xr + kk);
        v2f b;
        b[0] = W[(size_t)(kk + 0) * Ncols + col];
        b[1] = W[(size_t)(kk + 1) * Ncols + col];
        acc = __builtin_amdgcn_wmma_f32_16x16x4_f32(false, a, false, b,
                                                    (short)0, acc, false, false);
    }

    float bv = bias ? bias[col] : 0.0f;
#pragma unroll
    for (int r = 0; r < 8; ++r) {
        int m = (rt << 4) + r + (half << 3);
        float sc = rowscale ? rowscale[m] : 1.0f;
        Y[(size_t)m * Ncols + col] = acc[r] * sc + bv;
    }
}

// ---------------- BatchNorm stats: per-column sum / sumsq (width fixed 256) ----------------
#define BN_ROWS_PER_BLOCK 256
__global__ void bnstat_kernel(const float* __restrict__ x, float* __restrict__ sum,
                              float* __restrict__ sumsq, int Nrows) {
    int col = threadIdx.x;               // 256 threads, one column each (coalesced rows)
    int r0 = blockIdx.x * BN_ROWS_PER_BLOCK;
    int r1 = r0 + BN_ROWS_PER_BLOCK;
    if (r1 > Nrows) r1 = Nrows;
    float s = 0.0f, ss = 0.0f;
    for (int r = r0; r < r1; ++r) {
        float v = x[(size_t)r * 256 + col];
        s += v;
        ss += v * v;
    }
    atomAddF(&sum[col], s);
    atomAddF(&sumsq[col], ss);
}

__global__ void bnfinalize_kernel(const float* __restrict__ sum, const float* __restrict__ sumsq,
                                  const float* __restrict__ g, const float* __restrict__ beta,
                                  float* __restrict__ sA, float* __restrict__ sB, int Nrows) {
    int c = threadIdx.x;
    float invN = 1.0f / (float)Nrows;
    float mean = sum[c] * invN;
    float var  = sumsq[c] * invN - mean * mean;
    float inv  = rsqrtf(var + BN_EPS) * g[c];
    sA[c] = inv;
    sB[c] = beta[c] - mean * inv;
}

__global__ void bnrelu_kernel(float* __restrict__ x, const float* __restrict__ sA,
                              const float* __restrict__ sB, long long total) {
    long long i = (long long)blockIdx.x * blockDim.x + threadIdx.x;
    if (i < total) {
        int c = (int)(i & 255);
        x[i] = fmaxf(0.0f, x[i] * sA[c] + sB[c]);
    }
}

// ---------------- host ----------------
extern "C" void kernel_launch(void* const* d_in, const int* in_sizes, int n_in,
                              void* d_out, int out_size, void* d_ws, size_t ws_size,
                              hipStream_t stream) {
    const float* feat  = (const float*)d_in[0];
    const int*   src   = (const int*)d_in[1];
    const int*   dst   = (const int*)d_in[2];
    const float* W1    = (const float*)d_in[3];
    const float* b1    = (const float*)d_in[4];
    const float* g1    = (const float*)d_in[5];
    const float* beta1 = (const float*)d_in[6];
    const float* W2    = (const float*)d_in[7];
    const float* b2    = (const float*)d_in[8];
    const float* g2    = (const float*)d_in[9];
    const float* beta2 = (const float*)d_in[10];
    const float* Wl    = (const float*)d_in[11];
    const float* bl    = (const float*)d_in[12];

    const int D = 256, AOUT = 64;
    const int N = in_sizes[0] / D;
    const int E = in_sizes[1];

    // workspace carve-out
    char* ws = (char*)d_ws;
    size_t off = 0;
    auto carve = [&](size_t bytes) -> void* {
        void* p = ws + off;
        off += (bytes + 255) & ~(size_t)255;
        return p;
    };
    float* so      = (float*)carve((size_t)N * 4);          // rsqrt(out_deg)
    float* si      = (float*)carve((size_t)N * 4);          // rsqrt(in_deg)
    float* colsum  = (float*)carve(256 * 4);
    float* colsq   = (float*)carve(256 * 4);
    float* bnA     = (float*)carve(256 * 4);
    float* bnB     = (float*)carve(256 * 4);
    float* bufA    = (float*)carve((size_t)N * D * 4);
    float* bufB    = (float*)carve((size_t)N * D * 4);
    (void)ws_size; (void)n_in;

    const int T = 256;
    dim3 blk(T);
    long long aggThreads = (long long)E * 64;
    int aggBlocks   = (int)((aggThreads + T - 1) / T);
    int quadBlocks  = (int)(((long long)N * 64 + T - 1) / T);
    int ewBlocks    = (int)(((long long)N * 256 + T - 1) / T);
    int statBlocks  = (N + BN_ROWS_PER_BLOCK - 1) / BN_ROWS_PER_BLOCK;
    int gemmTiles16 = (N / 16) * (D / 16);     // 256-wide GEMMs
    int gemmBlocks16 = (gemmTiles16 + 7) / 8;  // 8 waves per 256-thread block
    int gemmTilesOut = (N / 16) * (AOUT / 16);
    int gemmBlocksOut = (gemmTilesOut + 7) / 8;

    // --- degrees -> rsqrt ---
    hipMemsetAsync(so, 0, (size_t)N * 4, stream);
    hipMemsetAsync(si, 0, (size_t)N * 4, stream);
    deg_kernel<<<(E + T - 1) / T, blk, 0, stream>>>(src, dst, so, si, E);
    rsqrt_deg_kernel<<<(N + T - 1) / T, blk, 0, stream>>>(so, si, N);

    // ===== layer 1 =====
    rowscale_kernel<<<quadBlocks, blk, 0, stream>>>(feat, so, bufA, N * 64);
    hipMemsetAsync(bufB, 0, (size_t)N * D * 4, stream);
    aggregate_kernel<<<aggBlocks, blk, 0, stream>>>(bufA, src, dst, bufB, E);
    wmma_gemm_kernel<<<gemmBlocks16, blk, 0, stream>>>(bufB, W1, b1, si, bufA, N, D, D);
    hipMemsetAsync(colsum, 0, 256 * 4, stream);
    hipMemsetAsync(colsq, 0, 256 * 4, stream);
    bnstat_kernel<<<statBlocks, blk, 0, stream>>>(bufA, colsum, colsq, N);
    bnfinalize_kernel<<<1, blk, 0, stream>>>(colsum, colsq, g1, beta1, bnA, bnB, N);
    bnrelu_kernel<<<ewBlocks, blk, 0, stream>>>(bufA, bnA, bnB, (long long)N * 256);

    // ===== layer 2 =====
    rowscale_kernel<<<quadBlocks, blk, 0, stream>>>(bufA, so, bufB, N * 64);
    hipMemsetAsync(bufA, 0, (size_t)N * D * 4, stream);
    aggregate_kernel<<<aggBlocks, blk, 0, stream>>>(bufB, src, dst, bufA, E);
    wmma_gemm_kernel<<<gemmBlocks16, blk, 0, stream>>>(bufA, W2, b2, si, bufB, N, D, D);
    hipMemsetAsync(colsum, 0, 256 * 4, stream);
    hipMemsetAsync(colsq, 0, 256 * 4, stream);
    bnstat_kernel<<<statBlocks, blk, 0, stream>>>(bufB, colsum, colsq, N);
    bnfinalize_kernel<<<1, blk, 0, stream>>>(colsum, colsq, g2, beta2, bnA, bnB, N);
    bnrelu_kernel<<<ewBlocks, blk, 0, stream>>>(bufB, bnA, bnB, (long long)N * 256);

    // ===== final linear 256 -> 64 =====
    wmma_gemm_kernel<<<gemmBlocksOut, blk, 0, stream>>>(bufB, Wl, bl, (const float*)nullptr,
                                                        (float*)d_out, N, D, AOUT);
}